// CustomizeL2Loss_87393994539075
// MI455X (gfx1250) — compile-verified
//
#include <hip/hip_runtime.h>

// Problem geometry: x,y are (N=256, C=3, T=600, V=25, M=2) f32, contiguous.
// Per sample: an 1800x50-float (== 1800x25-float2) row-major matrix.
#define N_SAMP 256
#define COLS 25                    // float2 columns; column j == joint v
#define ROWS 1800                  // C*T
#define STRIPS 36
#define ROWS_PER_STRIP 50          // 1800/36; divides 600 so t==0 only at k==0
#define THREADS (STRIPS * COLS)    // 900 threads per block
#define PLANE2 (ROWS * COLS)       // 45000 float2 per sample
#define TOTAL_ELEMS 23040000.0f    // 256*3*600*25*2

__global__ __launch_bounds__(THREADS)
void cl2_partial_kernel(const float2* __restrict__ x,
                        const float2* __restrict__ y,
                        float* __restrict__ per_n) {
    __shared__ float sA[THREADS];
    __shared__ float sB[THREADS];
    const int n   = blockIdx.x;
    const int tid = threadIdx.x;
    const int s   = tid / COLS;        // strip
    const int j   = tid % COLS;        // float2 column == joint v
    const float2* __restrict__ xb = x + (size_t)n * PLANE2;
    const float2* __restrict__ yb = y + (size_t)n * PLANE2;

    const int r0 = s * ROWS_PER_STRIP;
    int idx = r0 * COLS + j;

    // k == 0 (handles the c/t boundary; r0 % 600 == 0 iff s % 12 == 0)
    float2 xv = xb[idx];
    float2 yv = yb[idx];
    float dx = xv.x - yv.x, dy = xv.y - yv.y;
    float accB = dx * dx + dy * dy;
    float accA = 0.0f;
    if ((r0 % 600) != 0) {             // needs previous row from the strip above
        float2 xm = xb[idx - COLS];
        accA = fabsf(xv.x - xm.x) + fabsf(xv.y - xm.y);
    }
    float2 prev = xv;

    #pragma unroll 10
    for (int k = 1; k < ROWS_PER_STRIP; ++k) {
        idx += COLS;
        xv = xb[idx];
        yv = yb[idx];
        dx = xv.x - yv.x; dy = xv.y - yv.y;
        accB += dx * dx + dy * dy;                            // (x-y)^2 term
        accA += fabsf(xv.x - prev.x) + fabsf(xv.y - prev.y);  // |motion| term
        prev = xv;
    }

    // Deterministic fixed-order block reduction (no float atomics).
    sA[tid] = accA;
    sB[tid] = accB;
    __syncthreads();
    if (tid < COLS) {
        float a = 0.0f, b = 0.0f;
        for (int ss = 0; ss < STRIPS; ++ss) {
            a += sA[ss * COLS + tid];
            b += sB[ss * COLS + tid];
        }
        sA[tid] = a;
        sB[tid] = b;
    }
    __syncthreads();
    if (tid == 0) {
        // ratio[v] = 25*A[v]/sum_v A  (the 1/3594 mean factor cancels)
        float sumA = 0.0f, dot = 0.0f;
        for (int v = 0; v < COLS; ++v) {
            sumA += sA[v];
            dot  += sA[v] * sB[v];
        }
        per_n[n] = 25.0f * dot / sumA;   // sum_v ratio[v]*B[v]
    }
}

// ---- Final 256 -> 1 reduction via V_WMMA_F32_16X16X4_F32 ----------------
// D = ones(16x4) x B(4x16) + C  => every row of D holds the column sums of B.
// Four chained WMMAs consume the 256 per-sample losses; summing lane-resident
// D column sums over lanes 0..15 yields the exact grand total (the result is
// invariant to the B-operand row permutation since all columns are summed).
typedef float v2f_t __attribute__((ext_vector_type(2)));
typedef float v8f_t __attribute__((ext_vector_type(8)));

__global__ __launch_bounds__(32)
void cl2_final_reduce_wmma(const float* __restrict__ per_n,
                           float* __restrict__ out) {
    const int l   = threadIdx.x;           // 0..31: one full wave, EXEC all 1s
    const int col = l & 15;
    const int k0  = (l < 16) ? 0 : 2;      // assumed 4x16 B row interleave

    v2f_t a; a.x = 1.0f; a.y = 1.0f;       // all-ones A (16x4)
    v8f_t c = {};
    #pragma unroll
    for (int ch = 0; ch < 4; ++ch) {
        v2f_t b;
        b.x = per_n[ch * 64 + k0 * 16 + col];
        b.y = per_n[ch * 64 + (k0 + 1) * 16 + col];
        c = __builtin_amdgcn_wmma_f32_16x16x4_f32(
                /*neg_a=*/false, a, /*neg_b=*/false, b,
                /*c_mod=*/(short)0, c, /*reuse_a=*/false, /*reuse_b=*/false);
    }
    // C/D layout: lane l, VGPR0 = (M=0 or 8, N=l%16) => column sum S_{l%16}.
    float ssum = c[0];
    ssum += __shfl_xor(ssum, 1, 32);
    ssum += __shfl_xor(ssum, 2, 32);
    ssum += __shfl_xor(ssum, 4, 32);
    ssum += __shfl_xor(ssum, 8, 32);
    if (l == 0) out[0] = ssum * (1.0f / TOTAL_ELEMS);
}

extern "C" void kernel_launch(void* const* d_in, const int* in_sizes, int n_in,
                              void* d_out, int out_size, void* d_ws, size_t ws_size,
                              hipStream_t stream) {
    const float2* x = (const float2*)d_in[0];
    const float2* y = (const float2*)d_in[1];
    float* per_n = (float*)d_ws;   // 256 floats of scratch
    float* out   = (float*)d_out;

    cl2_partial_kernel<<<N_SAMP, THREADS, 0, stream>>>(x, y, per_n);
    cl2_final_reduce_wmma<<<1, 32, 0, stream>>>(per_n, out);
}